// DiffWarp_32787780338074
// MI455X (gfx1250) — compile-verified
//
#include <hip/hip_runtime.h>
#include <math.h>

#define HH 1080
#define WW 1920
#define TILES_PER_ROW (WW / 32)                  // 60
#define TOTAL_TILES ((HH * WW) / 32)             // 64800
#define TILES_PER_WAVE 4
#define WAVES_PER_BLOCK 8
#define NWAVES (TOTAL_TILES / TILES_PER_WAVE)    // 16200
#define NBLOCKS (NWAVES / WAVES_PER_BLOCK)       // 2025

typedef float v2f __attribute__((ext_vector_type(2)));
typedef float v8f __attribute__((ext_vector_type(8)));

// ---------------------------------------------------------------------------
// Setup (1 wave): every lane redundantly computes
//   M4 = [ K * T[:3,:3] * Kinv | K * T[:3,3] ]   with T = f1 @ inv(f2)
// (uniform lockstep work, costs the same as 1 lane), then writes its per-lane
// WMMA A-operand values (a1.xy for rows 0..2, a2.xy for rows 8..10) into a
// 32 x float4 table in the workspace. The main kernel needs nothing else:
//   q = M4 @ [d*x, d*y, d, 1],  gx = qx/qz, gy = qy/qz
// (cp[3] divide and normalize/denormalize round-trip cancel exactly).
// ---------------------------------------------------------------------------
__global__ void DiffWarp_setup(const float* __restrict__ f1,
                               const float* __restrict__ f2,
                               const float* __restrict__ K,
                               float4* __restrict__ tab) {
    const int lane = threadIdx.x;   // 0..31

    // ---- invert f2 (4x4), Gauss-Jordan with partial pivoting ----
    float a[4][8];
    for (int i = 0; i < 4; i++)
        for (int j = 0; j < 4; j++) {
            a[i][j]     = f2[i * 4 + j];
            a[i][j + 4] = (i == j) ? 1.0f : 0.0f;
        }
    for (int col = 0; col < 4; col++) {
        int piv = col;
        float best = fabsf(a[col][col]);
        for (int r = col + 1; r < 4; r++) {
            float v = fabsf(a[r][col]);
            if (v > best) { best = v; piv = r; }
        }
        if (piv != col)
            for (int j = 0; j < 8; j++) {
                float t = a[col][j]; a[col][j] = a[piv][j]; a[piv][j] = t;
            }
        float inv = 1.0f / a[col][col];
        for (int j = 0; j < 8; j++) a[col][j] *= inv;
        for (int r = 0; r < 4; r++)
            if (r != col) {
                float f = a[r][col];
                for (int j = 0; j < 8; j++) a[r][j] -= f * a[col][j];
            }
    }
    // ---- T = f1 @ inv(f2) ----
    float T[4][4];
    for (int i = 0; i < 4; i++)
        for (int j = 0; j < 4; j++) {
            float s = 0.0f;
            for (int k = 0; k < 4; k++) s += f1[i * 4 + k] * a[k][j + 4];
            T[i][j] = s;
        }
    // ---- Kinv (analytic adjugate) ----
    float k00 = K[0], k01 = K[1], k02 = K[2];
    float k10 = K[3], k11 = K[4], k12 = K[5];
    float k20 = K[6], k21 = K[7], k22 = K[8];
    float det = k00 * (k11 * k22 - k12 * k21)
              - k01 * (k10 * k22 - k12 * k20)
              + k02 * (k10 * k21 - k11 * k20);
    float id = 1.0f / det;
    float Ki[3][3] = {
        { (k11 * k22 - k12 * k21) * id, (k02 * k21 - k01 * k22) * id, (k01 * k12 - k02 * k11) * id },
        { (k12 * k20 - k10 * k22) * id, (k00 * k22 - k02 * k20) * id, (k02 * k10 - k00 * k12) * id },
        { (k10 * k21 - k11 * k20) * id, (k01 * k20 - k00 * k21) * id, (k00 * k11 - k01 * k10) * id }
    };
    // ---- M4 = [K @ T33 @ Kinv | K @ t] ----
    float U[3][3];
    for (int i = 0; i < 3; i++)
        for (int j = 0; j < 3; j++) {
            float s = 0.0f;
            for (int k = 0; k < 3; k++) s += T[i][k] * Ki[k][j];
            U[i][j] = s;
        }
    const float Km[3][3] = { {k00,k01,k02}, {k10,k11,k12}, {k20,k21,k22} };
    float M4[12];
    for (int i = 0; i < 3; i++) {
        for (int j = 0; j < 3; j++) {
            float s = 0.0f;
            for (int k = 0; k < 3; k++) s += Km[i][k] * U[k][j];
            M4[i * 4 + j] = s;
        }
        float s = 0.0f;
        for (int k = 0; k < 3; k++) s += Km[i][k] * T[k][3];
        M4[i * 4 + 3] = s;
    }

    // ---- per-lane A-matrix operand values ----
    // A layout (32-bit, 16x4): lane L<16 holds row M=L, K={0,1};
    //                          lane L>=16 holds row M=L-16, K={2,3}.
    const int  s  = lane & 15;
    const bool h  = lane >= 16;
    const float p0x = h ? M4[2]  : M4[0];
    const float p0y = h ? M4[3]  : M4[1];
    const float p1x = h ? M4[6]  : M4[4];
    const float p1y = h ? M4[7]  : M4[5];
    const float p2x = h ? M4[10] : M4[8];
    const float p2y = h ? M4[11] : M4[9];

    float4 t;
    // WMMA #1: M4 in rows 0..2  -> D rows 0..2 land in lanes 0-15
    t.x = (s == 0) ? p0x : ((s == 1) ? p1x : ((s == 2) ? p2x : 0.0f));
    t.y = (s == 0) ? p0y : ((s == 1) ? p1y : ((s == 2) ? p2y : 0.0f));
    // WMMA #2: M4 in rows 8..10 -> D rows 8..10 land in lanes 16-31
    t.z = (s == 8) ? p0x : ((s == 9) ? p1x : ((s == 10) ? p2x : 0.0f));
    t.w = (s == 8) ? p0y : ((s == 9) ? p1y : ((s == 10) ? p2y : 0.0f));
    tab[lane] = t;
}

// ---------------------------------------------------------------------------
// Main kernel: each wave processes TILES_PER_WAVE consecutive 32-pixel row
// tiles. Per tile, two V_WMMA_F32_16X16X4_F32 compute q = M4 @ [dx, dy, d, 1]
// for 16 pixels each; afterwards every lane owns its own pixel's (qx,qy,qz)
// in-register (no cross-lane traffic) and does the bilinear gather + planar
// store. Grid covers the domain exactly -> EXEC is all-ones for every WMMA.
// ---------------------------------------------------------------------------
__global__ __launch_bounds__(256) void DiffWarp_warp_kernel(
        const float* __restrict__ c1, const float* __restrict__ d1,
        const float4* __restrict__ tab, float* __restrict__ out) {
    const int lane = threadIdx.x & 31;
    const int wave = threadIdx.x >> 5;
    const int wglobal = blockIdx.x * WAVES_PER_BLOCK + wave;   // [0, 16200)
    const int tile0 = wglobal * TILES_PER_WAVE;

    const int  s = lane & 15;
    const bool h = (lane >> 4) != 0;

    // loop-invariant WMMA A operands (one b128 load, built by setup kernel)
    const float4 t = tab[lane];
    v2f a1; a1.x = t.x; a1.y = t.y;
    v2f a2; a2.x = t.z; a2.y = t.w;

#pragma unroll
    for (int it = 0; it < TILES_PER_WAVE; ++it) {
        const int tile = tile0 + it;
        const int r  = tile / TILES_PER_ROW;
        const int c0 = (tile % TILES_PER_ROW) * 32;

        // prefetch next tile's depth line (clamped; speculative-safe anyway)
        {
            const int tn = (tile + 1 < TOTAL_TILES) ? (tile + 1) : tile;
            const int rn  = tn / TILES_PER_ROW;
            const int cn  = (tn % TILES_PER_ROW) * 32;
            __builtin_prefetch(d1 + (size_t)rn * WW + cn, 0, 3);
        }

        const float y  = (float)r;
        const float xA = (float)(c0 + s);
        const float xB = (float)(c0 + 16 + s);
        const float dA = d1[r * WW + c0 + s];
        const float dB = d1[r * WW + c0 + 16 + s];

        // B operand (4x16): lane n<16 supplies rows K=0,1 = (d*x, d*y);
        //                   lane n+16 supplies rows K=2,3 = (d, 1)
        v2f b1, b2;
        b1.x = h ? dA   : dA * xA;
        b1.y = h ? 1.0f : dA * y;
        b2.x = h ? dB   : dB * xB;
        b2.y = h ? 1.0f : dB * y;

        v8f acc1 = {0.f, 0.f, 0.f, 0.f, 0.f, 0.f, 0.f, 0.f};
        v8f acc2 = {0.f, 0.f, 0.f, 0.f, 0.f, 0.f, 0.f, 0.f};
        acc1 = __builtin_amdgcn_wmma_f32_16x16x4_f32(false, a1, false, b1,
                                                     (short)0, acc1, false, false);
        acc2 = __builtin_amdgcn_wmma_f32_16x16x4_f32(false, a2, false, b2,
                                                     (short)0, acc2, false, false);

        // per-lane projected homogeneous point
        const float qx = h ? acc2[0] : acc1[0];
        const float qy = h ? acc2[1] : acc1[1];
        const float qz = h ? acc2[2] : acc1[2];
        const float gx = qx / qz;
        const float gy = qy / qz;

        // bilinear sample, zeros padding, align_corners semantics
        const float x0f = floorf(gx), y0f = floorf(gy);
        const float wx1 = gx - x0f, wy1 = gy - y0f;
        const float wx0 = 1.0f - wx1, wy0 = 1.0f - wy1;

        float o0 = 0.f, o1 = 0.f, o2 = 0.f;
#pragma unroll
        for (int tp = 0; tp < 4; tp++) {
            const float xf = x0f + (float)(tp & 1);
            const float yf = y0f + (float)(tp >> 1);
            const float w  = ((tp & 1) ? wx1 : wx0) * ((tp >> 1) ? wy1 : wy0);
            const bool valid = (xf >= 0.f) && (xf <= (float)(WW - 1)) &&
                               (yf >= 0.f) && (yf <= (float)(HH - 1));
            const int xi = (int)fminf(fmaxf(xf, 0.f), (float)(WW - 1));
            const int yi = (int)fminf(fmaxf(yf, 0.f), (float)(HH - 1));
            const float* p = c1 + ((size_t)yi * WW + xi) * 3;
            const float v0 = p[0], v1 = p[1], v2 = p[2];
            o0 += w * (valid ? v0 : 0.f);
            o1 += w * (valid ? v1 : 0.f);
            o2 += w * (valid ? v2 : 0.f);
        }

        const int col = c0 + lane;                  // lane-to-pixel map is exact
        const size_t base = (size_t)r * WW + col;
        out[base]                       = o0;       // plane 0
        out[(size_t)HH * WW + base]     = o1;       // plane 1
        out[2 * (size_t)HH * WW + base] = o2;       // plane 2
    }
}

extern "C" void kernel_launch(void* const* d_in, const int* in_sizes, int n_in,
                              void* d_out, int out_size, void* d_ws, size_t ws_size,
                              hipStream_t stream) {
    const float* f1 = (const float*)d_in[0];   // f1_pose (16)
    const float* f2 = (const float*)d_in[1];   // f2_pose (16)
    const float* K  = (const float*)d_in[2];   // K (9)
    const float* c1 = (const float*)d_in[3];   // (H,W,3)
    const float* d1 = (const float*)d_in[4];   // (H,W)
    float*  out = (float*)d_out;               // (1,3,H,W)
    float4* tab = (float4*)d_ws;               // 32 x float4 per-lane A operands

    DiffWarp_setup<<<1, 32, 0, stream>>>(f1, f2, K, tab);
    DiffWarp_warp_kernel<<<NBLOCKS, 256, 0, stream>>>(c1, d1, tab, out);
}